// RZ_50259707298063
// MI455X (gfx1250) — compile-verified
//
#include <hip/hip_runtime.h>
#include <math.h>

// RZ on 12 qubits (DIM=2, J=1): U is DIAGONAL. y[k,b] = e^{-iA(k)} * x[k,b],
// A(k) = 0.5 * sum_i sigma_i(k) * theta_i, sigma = +1 if bit(11-i) of k is 0.
// Pure bandwidth-bound streaming kernel (~33.5 MB -> ~1.4 us at 23.3 TB/s).
// Data movement uses the CDNA5 async copy engine (GLOBAL_LOAD_ASYNC_TO_LDS_B128,
// ASYNCcnt) with LDS double buffering; stores are interleaved-complex B128.

namespace {
constexpr int kWires   = 12;
constexpr int kD       = 1 << kWires;   // 4096 rows
constexpr int kBatch   = 512;           // columns
constexpr int kThreads = 128;           // 4 wave32 per block
constexpr int kNB4     = kBatch / 4;    // 128 float4 per row
constexpr int kRows    = 2;             // rows per block (double buffer)
}

// Async global->LDS copy of 16 bytes per lane. GV addressing: 64-bit VGPR
// address, saddr = off. VDST VGPR carries the LDS byte address; the low 32
// bits of a flat shared pointer are exactly that offset (aperture is in the
// high 32 bits per the CDNA5 aperture check). Verified in disasm: offsets
// folded to pure LDS byte addresses.
__device__ __forceinline__ void async_copy_b128(const float4* __restrict__ g,
                                                float4* l) {
  unsigned lds_off = (unsigned)(unsigned long long)l;
  unsigned long long gaddr = (unsigned long long)g;
  asm volatile("global_load_async_to_lds_b128 %0, %1, off"
               :: "v"(lds_off), "v"(gaddr)
               : "memory");
}

template <int N>
__device__ __forceinline__ void wait_asynccnt() {
  asm volatile("s_wait_asynccnt %0" :: "n"(N) : "memory");
}

__global__ __launch_bounds__(kThreads) void rz12_diag_phase_kernel(
    const float* __restrict__ xr, const float* __restrict__ xi,
    const float* __restrict__ ang, float* __restrict__ out) {
  __shared__ float4 bR[kRows][kThreads];   // 2 x 2KB
  __shared__ float4 bI[kRows][kThreads];   // 2 x 2KB
  const int t  = threadIdx.x;
  const int k0 = blockIdx.x * kRows;       // even row index; bit0(k) == r

  const float4* xr4 = (const float4*)xr;
  const float4* xi4 = (const float4*)xi;

  // Fill both stages up front: 4 outstanding async B128 ops per wave (2 KB
  // in flight per wave; 8192 waves machine-wide ~= 16 MB in flight, sized
  // for 23.3 TB/s x ~600ns HBM latency by Little's law).
  async_copy_b128(xr4 + (size_t)(k0 + 0) * kNB4 + t, &bR[0][t]);
  async_copy_b128(xi4 + (size_t)(k0 + 0) * kNB4 + t, &bI[0][t]);
  async_copy_b128(xr4 + (size_t)(k0 + 1) * kNB4 + t, &bR[1][t]);
  async_copy_b128(xi4 + (size_t)(k0 + 1) * kNB4 + t, &bI[1][t]);

  // Angle sum for the fixed high 11 bits of k; only wire 11 (bit 0) varies
  // between this block's two rows. Uniform -> pure SALU select chain.
  float th11  = ang[kWires - 1];
  float Abase = 0.0f;
#pragma unroll
  for (int i = 0; i < kWires - 1; ++i) {
    float th = ang[i];
    Abase += ((k0 >> (kWires - 1 - i)) & 1) ? -th : th;
  }

#pragma unroll
  for (int r = 0; r < kRows; ++r) {
    // Wait for this stage's data (async ops complete in issue order).
    if (r == 0) wait_asynccnt<2>(); else wait_asynccnt<0>();

    float4 vr = bR[r][t];
    float4 vi = bI[r][t];

    float A = 0.5f * (Abase + (r ? -th11 : th11));
    // |A| is a few radians at most (angles ~ N(0,1)); hardware sin/cos
    // (TRANS32, co-executes with VALU) is plenty accurate and avoids the
    // ~400-instruction Payne-Hanek slow path of precise sincosf.
    float s, c;
    __sincosf(A, &s, &c);                 // phase = c - i*s

    // y = (c - i s)(xr + i xi): re = c*xr + s*xi, im = c*xi - s*xr
    float4 o0, o1;
    o0.x = fmaf(c, vr.x,  s * vi.x);  o0.y = fmaf(c, vi.x, -s * vr.x);
    o0.z = fmaf(c, vr.y,  s * vi.y);  o0.w = fmaf(c, vi.y, -s * vr.y);
    o1.x = fmaf(c, vr.z,  s * vi.z);  o1.y = fmaf(c, vi.z, -s * vr.z);
    o1.z = fmaf(c, vr.w,  s * vi.w);  o1.w = fmaf(c, vi.w, -s * vr.w);

    // complex64 [D, B] flattened as float32 pairs: (re, im) interleaved.
    float* ob = out + ((size_t)(k0 + r) * kBatch + (size_t)t * 4) * 2;
    ((float4*)ob)[0] = o0;                // B128 store
    ((float4*)ob)[1] = o1;                // B128 store
  }
}

extern "C" void kernel_launch(void* const* d_in, const int* in_sizes, int n_in,
                              void* d_out, int out_size, void* d_ws, size_t ws_size,
                              hipStream_t stream) {
  (void)in_sizes; (void)n_in; (void)out_size; (void)d_ws; (void)ws_size;
  const float* xr  = (const float*)d_in[0];   // x_real [D, B]
  const float* xi  = (const float*)d_in[1];   // x_imag [D, B]
  const float* ang = (const float*)d_in[2];   // angle  [12]
  float*       out = (float*)d_out;           // complex64 [D, B] as float pairs

  dim3 grid(kD / kRows);                      // 2048 blocks
  dim3 block(kThreads);                       // 128 threads = 4 wave32
  rz12_diag_phase_kernel<<<grid, block, 0, stream>>>(xr, xi, ang, out);
}